// GameOutcomePredictorV2_33784212750366
// MI455X (gfx1250) — compile-verified
//
#include <hip/hip_runtime.h>
#include <hip/hip_bf16.h>
#include <math.h>

typedef float v2f __attribute__((ext_vector_type(2)));
typedef float v8f __attribute__((ext_vector_type(8)));

#define NEG_SLOPE 0.2f

// ---------------- helpers: order-preserving float <-> uint encoding for atomic max --
__device__ __forceinline__ unsigned fenc(float f) {
  unsigned u = __float_as_uint(f);
  return (u & 0x80000000u) ? ~u : (u | 0x80000000u);
}
__device__ __forceinline__ float fdec(unsigned e) {
  return (e & 0x80000000u) ? __uint_as_float(e ^ 0x80000000u)
                           : __uint_as_float(~e);
}

// ---------------- zero-pad a 2D matrix into workspace -------------------------------
__global__ void pad2d(const float* __restrict__ src, float* __restrict__ dst,
                      int Rs, int Cs, int Rd, int Cd) {
  long i = (long)blockIdx.x * blockDim.x + threadIdx.x;
  if (i >= (long)Rd * Cd) return;
  int r = (int)(i / Cd), c = (int)(i % Cd);
  dst[i] = (r < Rs && c < Cs) ? src[(long)r * Cs + c] : 0.0f;
}

// ---------------- fp32 WMMA GEMM with LDS-staged B ----------------------------------
// Requirements: K % 4 == 0, N % (16*NT) == 0.  Block = 256 threads (8 waves).
// Each wave computes a 16 x (16*NT) strip of C: one A frag feeds NT WMMAs per K-step.
// B is staged in LDS *pair-packed*: element (k,n) lives at smem[(k>>1)*2N + 2n + (k&1)],
// so a lane's B frag (rows ka, ka+1 at column n) is one aligned ds_load_b64.
template <int NT>
__global__ __launch_bounds__(256) void wmma_gemm_lds(const float* __restrict__ A,
                                                     const float* __restrict__ B,
                                                     float* __restrict__ C,
                                                     int M, int N, int K) {
  __shared__ float smem[8192];                 // 32KB B staging buffer
  const int tid  = threadIdx.x;
  const int wave = tid >> 5;
  const int lane = tid & 31;
  const int half = lane >> 4;                  // 0: K pair {0,1}, 1: {2,3}
  const int l15  = lane & 15;

  const int tile_m = blockIdx.x * 8 + wave;
  const int row    = tile_m * 16 + l15;
  const int rowc   = row < M ? row : M - 1;    // clamp loads; stores are guarded
  const int nbase  = blockIdx.y * (16 * NT);   // first column of this block's strip

  const int KC = (8192 / N) & ~3;              // K-chunk that fits in smem, mult of 4

  v8f acc[NT] = {};
  for (int kc = 0; kc < K; kc += KC) {
    const int kl = (K - kc) < KC ? (K - kc) : KC;
    __syncthreads();
    for (int i = tid; i < kl * N; i += 256) {
      const int k = i / N, n = i % N;
      smem[((k >> 1) * N + n) * 2 + (k & 1)] = B[(long)kc * N + i];
    }
    __syncthreads();

    const float* Arow = A + (long)rowc * K + kc;
    for (int k0 = 0; k0 < kl; k0 += 4) {
      const int ka = k0 + half * 2;            // even
      v2f a = *(const v2f*)(Arow + ka);        // one unconditional b64 load
      const float* brow = &smem[((ka >> 1) * N + nbase + l15) * 2];
#pragma unroll
      for (int j = 0; j < NT; ++j) {
        v2f b = *(const v2f*)(brow + j * 32);  // one aligned ds_load_b64
        acc[j] = __builtin_amdgcn_wmma_f32_16x16x4_f32(
            false, a, false, b, (short)0, acc[j], false, false);
      }
    }
  }

#pragma unroll
  for (int j = 0; j < NT; ++j) {
#pragma unroll
    for (int r = 0; r < 8; ++r) {
      const int orow = tile_m * 16 + r + half * 8;
      if (orow < M) C[(long)orow * N + nbase + j * 16 + l15] = acc[j][r];
    }
  }
}

// ---------------- init kernels (graph-capture safe; no hipMemset) -------------------
__global__ void init_f32(float* p, long n, float v) {
  long i = (long)blockIdx.x * blockDim.x + threadIdx.x;
  if (i < n) p[i] = v;
}
__global__ void init_u32(unsigned* p, long n, unsigned v) {
  long i = (long)blockIdx.x * blockDim.x + threadIdx.x;
  if (i < n) p[i] = v;
}

// ---------------- per-node attention logits ----------------------------------------
__global__ void alpha_kernel(const float* __restrict__ H,
                             const float* __restrict__ a_src,
                             const float* __restrict__ a_dst,
                             float* __restrict__ as, float* __restrict__ ad,
                             int N, int heads, int C) {
  int i = blockIdx.x * blockDim.x + threadIdx.x;
  if (i >= N * heads) return;
  int n = i / heads, h = i % heads;
  const float* hp = H + (long)n * heads * C + (long)h * C;
  float ss = 0.f, sd = 0.f;
  for (int c = 0; c < C; ++c) {
    float v = hp[c];
    ss += v * a_src[h * C + c];
    sd += v * a_dst[h * C + c];
  }
  as[i] = ss;
  ad[i] = sd;
}

// ---------------- edge pass 1: leaky-relu logit + segment max (atomic) --------------
__global__ void edge_max(const int* __restrict__ ei, int E, int N, int heads,
                         const float* __restrict__ as, const float* __restrict__ ad,
                         float* __restrict__ eval, unsigned* __restrict__ m) {
  long i = (long)blockIdx.x * blockDim.x + threadIdx.x;
  long ET = (long)E + N;
  if (i >= ET * heads) return;
  long e = i / heads;
  int h = (int)(i % heads);
  int s = (e < E) ? ei[e] : (int)(e - E);
  int d = (e < E) ? ei[(long)E + e] : (int)(e - E);
  float v = as[(long)s * heads + h] + ad[(long)d * heads + h];
  v = (v > 0.f) ? v : NEG_SLOPE * v;
  eval[i] = v;
  atomicMax(&m[(long)d * heads + h], fenc(v));
}

// ---------------- edge pass 2: exp(e - max) + segment sum (atomic) ------------------
__global__ void edge_expsum(const int* __restrict__ ei, int E, int N, int heads,
                            const unsigned* __restrict__ m,
                            float* __restrict__ eval, float* __restrict__ ssum) {
  long i = (long)blockIdx.x * blockDim.x + threadIdx.x;
  long ET = (long)E + N;
  if (i >= ET * heads) return;
  long e = i / heads;
  int h = (int)(i % heads);
  int d = (e < E) ? ei[(long)E + e] : (int)(e - E);
  float ee = __expf(eval[i] - fdec(m[(long)d * heads + h]));
  eval[i] = ee;
  atomicAdd(&ssum[(long)d * heads + h], ee);
}

// ---------------- edge pass 3: weighted message aggregation -------------------------
__global__ void edge_agg(const int* __restrict__ ei, int E, int N, int heads, int C,
                         const float* __restrict__ H,
                         const float* __restrict__ eval,
                         const float* __restrict__ ssum,
                         float* __restrict__ agg) {
  long e = blockIdx.x;
  int t = threadIdx.x;            // 0 .. heads*C-1
  int h = t / C;
  int s = (e < E) ? ei[e] : (int)(e - E);
  int d = (e < E) ? ei[(long)E + e] : (int)(e - E);
  float w = eval[e * heads + h] / ssum[(long)d * heads + h];
  atomicAdd(&agg[(long)d * heads * C + t], H[(long)s * heads * C + t] * w);
}

// ---------------- bias + ELU (in place) ---------------------------------------------
__global__ void bias_elu(float* __restrict__ x, const float* __restrict__ b,
                         long n, int F) {
  long i = (long)blockIdx.x * blockDim.x + threadIdx.x;
  if (i >= n) return;
  float v = x[i] + b[i % F];
  x[i] = (v > 0.f) ? v : (__expf(v) - 1.f);
}

// ---------------- global mean pool (atomic sums + counts) ---------------------------
__global__ void pool_kernel(const float* __restrict__ h, const int* __restrict__ batch,
                            float* __restrict__ psum, float* __restrict__ pcnt,
                            int N, int C) {
  int n = blockIdx.x;
  int c = threadIdx.x;
  int g = batch[n];
  atomicAdd(&psum[(long)g * C + c], h[(long)n * C + c]);
  if (c == 0) atomicAdd(&pcnt[g], 1.0f);
}

// ---------------- final MLP: relu(pool@fc1+b)@fc2+b ---------------------------------
__global__ void mlp_kernel(const float* __restrict__ psum, const float* __restrict__ pcnt,
                           const float* __restrict__ w1, const float* __restrict__ b1,
                           const float* __restrict__ w2, const float* __restrict__ b2,
                           float* __restrict__ out, int G) {
  __shared__ float pooled[64];
  __shared__ float hid[64];
  int g = blockIdx.x;
  int c = threadIdx.x;           // blockDim = 64
  float cnt = fmaxf(pcnt[g], 1.0f);
  pooled[c] = psum[(long)g * 64 + c] / cnt;
  __syncthreads();
  float acc = b1[c];
  for (int k = 0; k < 64; ++k) acc += pooled[k] * w1[k * 64 + c];
  hid[c] = fmaxf(acc, 0.0f);
  __syncthreads();
  if (c < 3) {
    float o = b2[c];
    for (int k = 0; k < 64; ++k) o += hid[k] * w2[k * 3 + c];
    out[(long)g * 3 + c] = o;
  }
}

// ====================================================================================
extern "C" void kernel_launch(void* const* d_in, const int* in_sizes, int n_in,
                              void* d_out, int out_size, void* d_ws, size_t ws_size,
                              hipStream_t stream) {
  const int IN = 22, INP = 24, HID = 64, HEADS = 4;
  const float* x      = (const float*)d_in[0];
  const int*   ei     = (const int*)d_in[1];
  const int*   batch  = (const int*)d_in[2];
  const float* W1     = (const float*)d_in[3];
  const float* a_src1 = (const float*)d_in[4];
  const float* a_dst1 = (const float*)d_in[5];
  const float* b1     = (const float*)d_in[6];
  const float* W2     = (const float*)d_in[7];
  const float* a_src2 = (const float*)d_in[8];
  const float* a_dst2 = (const float*)d_in[9];
  const float* b2     = (const float*)d_in[10];
  const float* fc1_w  = (const float*)d_in[11];
  const float* fc1_b  = (const float*)d_in[12];
  const float* fc2_w  = (const float*)d_in[13];
  const float* fc2_b  = (const float*)d_in[14];
  float* out = (float*)d_out;

  const int N  = in_sizes[0] / IN;       // 50000
  const int E  = in_sizes[1] / 2;        // 800000
  const int G  = out_size / 3;           // 2048
  const long ET = (long)E + N;           // edges incl. self-loops
  const int F1 = HEADS * HID;            // 256

  // carve workspace
  char* cur = (char*)d_ws;
  auto alloc = [&](long bytes) {
    char* p = cur;
    cur += (bytes + 255) & ~255L;
    return (void*)p;
  };
  float*    xpad  = (float*)alloc((long)N * INP * 4);     // x padded to K=24
  float*    W1pad = (float*)alloc((long)INP * F1 * 4);    // W1 padded to K=24
  float*    H1    = (float*)alloc((long)N * F1 * 4);
  float*    as1   = (float*)alloc((long)N * HEADS * 4);
  float*    ad1   = (float*)alloc((long)N * HEADS * 4);
  unsigned* m1    = (unsigned*)alloc((long)N * HEADS * 4);
  float*    s1    = (float*)alloc((long)N * HEADS * 4);
  float*    ev1   = (float*)alloc(ET * HEADS * 4);
  float*    agg1  = (float*)alloc((long)N * F1 * 4);      // becomes layer-1 output
  float*    H2    = (float*)alloc((long)N * HID * 4);
  float*    as2   = (float*)alloc((long)N * 4);
  float*    ad2   = (float*)alloc((long)N * 4);
  unsigned* m2    = (unsigned*)alloc((long)N * 4);
  float*    s2    = (float*)alloc((long)N * 4);
  float*    ev2   = (float*)alloc(ET * 4);
  float*    agg2  = (float*)alloc((long)N * HID * 4);     // becomes layer-2 output
  float*    psum  = (float*)alloc((long)G * HID * 4);
  float*    pcnt  = (float*)alloc((long)G * 4);
  (void)ws_size;

  auto gridN = [](long n, int b) { return (unsigned)((n + b - 1) / b); };
  const unsigned mblocks = (unsigned)(((N + 15) / 16 + 7) / 8);  // 8 waves/block

  // ---- layer 1 -------------------------------------------------------------------
  pad2d<<<gridN((long)N * INP, 256), 256, 0, stream>>>(x, xpad, N, IN, N, INP);
  pad2d<<<gridN((long)INP * F1, 256), 256, 0, stream>>>(W1, W1pad, IN, F1, INP, F1);
  {
    dim3 g(mblocks, F1 / 64);   // NT=4 -> 64 cols per wave strip
    wmma_gemm_lds<4><<<g, 256, 0, stream>>>(xpad, W1pad, H1, N, F1, INP);
  }
  alpha_kernel<<<gridN((long)N * HEADS, 256), 256, 0, stream>>>(H1, a_src1, a_dst1,
                                                                as1, ad1, N, HEADS, HID);
  init_u32<<<gridN((long)N * HEADS, 256), 256, 0, stream>>>(m1, (long)N * HEADS, 0u);
  init_f32<<<gridN((long)N * HEADS, 256), 256, 0, stream>>>(s1, (long)N * HEADS, 0.f);
  init_f32<<<gridN((long)N * F1, 256), 256, 0, stream>>>(agg1, (long)N * F1, 0.f);
  edge_max<<<gridN(ET * HEADS, 256), 256, 0, stream>>>(ei, E, N, HEADS, as1, ad1, ev1, m1);
  edge_expsum<<<gridN(ET * HEADS, 256), 256, 0, stream>>>(ei, E, N, HEADS, m1, ev1, s1);
  edge_agg<<<(unsigned)ET, HEADS * HID, 0, stream>>>(ei, E, N, HEADS, HID, H1, ev1, s1, agg1);
  bias_elu<<<gridN((long)N * F1, 256), 256, 0, stream>>>(agg1, b1, (long)N * F1, F1);

  // ---- layer 2 (heads = 1, K=256 already aligned) --------------------------------
  {
    dim3 g(mblocks, HID / 64);  // N=64 -> one 64-col strip
    wmma_gemm_lds<4><<<g, 256, 0, stream>>>(agg1, W2, H2, N, HID, F1);
  }
  alpha_kernel<<<gridN((long)N, 256), 256, 0, stream>>>(H2, a_src2, a_dst2,
                                                        as2, ad2, N, 1, HID);
  init_u32<<<gridN((long)N, 256), 256, 0, stream>>>(m2, (long)N, 0u);
  init_f32<<<gridN((long)N, 256), 256, 0, stream>>>(s2, (long)N, 0.f);
  init_f32<<<gridN((long)N * HID, 256), 256, 0, stream>>>(agg2, (long)N * HID, 0.f);
  edge_max<<<gridN(ET, 256), 256, 0, stream>>>(ei, E, N, 1, as2, ad2, ev2, m2);
  edge_expsum<<<gridN(ET, 256), 256, 0, stream>>>(ei, E, N, 1, m2, ev2, s2);
  edge_agg<<<(unsigned)ET, HID, 0, stream>>>(ei, E, N, 1, HID, H2, ev2, s2, agg2);
  bias_elu<<<gridN((long)N * HID, 256), 256, 0, stream>>>(agg2, b2, (long)N * HID, HID);

  // ---- pooling + MLP head --------------------------------------------------------
  init_f32<<<gridN((long)G * HID, 256), 256, 0, stream>>>(psum, (long)G * HID, 0.f);
  init_f32<<<gridN((long)G, 256), 256, 0, stream>>>(pcnt, (long)G, 0.f);
  pool_kernel<<<N, HID, 0, stream>>>(agg2, batch, psum, pcnt, N, HID);
  mlp_kernel<<<G, HID, 0, stream>>>(psum, pcnt, fc1_w, fc1_b, fc2_w, fc2_b, out, G);
}